// GraphConv_21423296872853
// MI455X (gfx1250) — compile-verified
//
#include <hip/hip_runtime.h>
#include <stdint.h>

// GraphConv scatter-gather for MI455X (gfx1250, wave32).
// Memory-bound: vertex table (25.6MB) + output (25.6MB) are L2-resident;
// edge stream comes from HBM. Strategy: async-stage edge data (low dwords of
// the int64 indices only -> half the traffic) into LDS, b128 gathers,
// hardware f32 atomics for scatter, prefetch to hide L2 latency.
// All index/offset math is 32-bit: ids < 50000, byte offsets < 26MB.

#define N_FEAT   128
#define CHUNK    256     // edges per block
#define BLOCK    256     // 8 wave32s

typedef float f32x4 __attribute__((ext_vector_type(4)));

// ---------- helpers -------------------------------------------------------

__device__ __forceinline__ void atom_add_f32(float* p, float v) {
#if __has_builtin(__builtin_amdgcn_global_atomic_fadd_f32)
    __builtin_amdgcn_global_atomic_fadd_f32(
        (__attribute__((address_space(1))) float*)p, v);
#else
    __hip_atomic_fetch_add(p, v, __ATOMIC_RELAXED, __HIP_MEMORY_SCOPE_AGENT);
#endif
}

__device__ __forceinline__ void async_ld_b32(const void* gsrc, void* ldst) {
#if __has_builtin(__builtin_amdgcn_global_load_async_to_lds_b32)
    __builtin_amdgcn_global_load_async_to_lds_b32(
        (__attribute__((address_space(1))) int*)gsrc,
        (__attribute__((address_space(3))) int*)ldst, 0, 0);
#else
    *(int*)ldst = *(const int*)gsrc;   // sync fallback
#endif
}

__device__ __forceinline__ void wait_async0() {
#if __has_builtin(__builtin_amdgcn_s_wait_asynccnt)
    __builtin_amdgcn_s_wait_asynccnt(0);
#else
    asm volatile("s_wait_asynccnt 0" ::: "memory");
#endif
}

// ---------- probe kernel (FIRST in module so disasm snippet shows the
// ---------- lowering of the f32 atomic and the prefetch; never launched) ---

__global__ void atomic_probe_kernel(float* p, float v) {
    __builtin_prefetch(p + threadIdx.x + 64, 0, 1);
    atom_add_f32(p + threadIdx.x, v);
}

// ---------- kernels -------------------------------------------------------

__global__ __launch_bounds__(BLOCK)
void edge_scatter_kernel(const float* __restrict__ vin,
                         const int* __restrict__ eidx32, // int64 view as 2x i32
                         const float* __restrict__ enorm,
                         const float* __restrict__ esgn,
                         float* __restrict__ out,
                         int n_edges) {
    __shared__ int   lds_sidx[CHUNK];
    __shared__ int   lds_tidx[CHUNK];
    __shared__ float lds_w[CHUNK];

    const int t    = threadIdx.x;
    const int cbase = blockIdx.x * CHUNK;
    int e = cbase + t;
    const int valid_in_chunk = min(CHUNK, n_edges - cbase);
    if (e >= n_edges) e = n_edges - 1;        // clamp staging loads (benign dup)

    // Stage this block's edge data. Indices are int64 but values < 50000, so
    // async-DMA only the low dword of each: sidx64[e] -> eidx32[2e],
    // tidx64[e] -> eidx32[2(n_edges+e)].
    async_ld_b32(eidx32 + 2 * e, &lds_sidx[t]);
    async_ld_b32(eidx32 + 2 * (n_edges + e), &lds_tidx[t]);
    lds_w[t] = enorm[e] * esgn[e];
    wait_async0();
    __syncthreads();

    const int lane = t & 31;
    const int wave = t >> 5;
    const int col  = lane * 4;                // 32 lanes x float4 = 128 feats

    const int base = wave * 32;
    const int nk   = min(32, valid_in_chunk - base);
    for (int k = 0; k < nk; ++k) {
        const int slot = base + k;
        const int s = lds_sidx[slot];         // LDS broadcast reads
        const int d = lds_tidx[slot];
        const float w = lds_w[slot];

        // Unconditional prefetch of the next edge's source row (clamped slot)
        const int sn = lds_sidx[min(slot + 1, CHUNK - 1)];
        __builtin_prefetch(vin + sn * N_FEAT + col, 0, 1);

        const f32x4 v = *(const f32x4*)(vin + s * N_FEAT + col);
        float* op = out + d * N_FEAT + col;
        atom_add_f32(op + 0, v.x * w);
        atom_add_f32(op + 1, v.y * w);
        atom_add_f32(op + 2, v.z * w);
        atom_add_f32(op + 3, v.w * w);
    }
}

__global__ __launch_bounds__(BLOCK)
void zero_out_kernel(float* __restrict__ out, int n4) {
    int i = blockIdx.x * BLOCK + threadIdx.x;
    if (i < n4) {
        f32x4 z = {0.f, 0.f, 0.f, 0.f};
        ((f32x4*)out)[i] = z;
    }
}

// ---------- launch --------------------------------------------------------

extern "C" void kernel_launch(void* const* d_in, const int* in_sizes, int n_in,
                              void* d_out, int out_size, void* d_ws, size_t ws_size,
                              hipStream_t stream) {
    const float* vin    = (const float*)d_in[0];
    const int*   eidx32 = (const int*)d_in[1];   // int64 (2, E) viewed as i32
    const float* enorm  = (const float*)d_in[2];
    const float* esgn   = (const float*)d_in[3];
    float*       out    = (float*)d_out;

    const int n_edges = in_sizes[2];          // enorm has one entry per edge
    const int n4      = out_size / 4;         // out is float32, 128-feat rows

    zero_out_kernel<<<(n4 + BLOCK - 1) / BLOCK, BLOCK, 0, stream>>>(out, n4);

    const int nblocks = (n_edges + CHUNK - 1) / CHUNK;    // 2500 for 640k edges
    edge_scatter_kernel<<<nblocks, BLOCK, 0, stream>>>(
        vin, eidx32, enorm, esgn, out, n_edges);
}